// LvqLinear_9689446220465
// MI455X (gfx1250) — compile-verified
//
#include <hip/hip_runtime.h>
#include <hip/hip_bf16.h>

typedef __attribute__((ext_vector_type(2))) float v2f;
typedef __attribute__((ext_vector_type(4))) float v4f;
typedef __attribute__((ext_vector_type(8))) float v8f;

#define O_DIM 2048
#define I_DIM 2048
#define M_DIM 2048        // 4 * 512
#define VEC 8
#define NPOS 256          // I / VEC
#define NCODE 256
#define NGROUP 16         // I / 128

// ---------------------------------------------------------------------------
// Phase 1: residual vector quantization of the weight matrix.
// One block = one position p (8 input columns) x 256 output rows.
// Codebooks (both levels) + precomputed ||c||^2 staged in LDS.
// Writes qwT[i][o] (transposed, rescaled) into workspace for coalesced
// K-major B loads in the GEMM.
// ---------------------------------------------------------------------------
__global__ __launch_bounds__(256) void rvq_quant_kernel(
    const float* __restrict__ weight,     // [O, I]
    const float* __restrict__ scales,     // [O, I/128]
    const float* __restrict__ codebooks,  // [2, NPOS, NCODE, VEC]
    float* __restrict__ qwT)              // [I, O]
{
  __shared__ float cb0[NCODE * VEC];
  __shared__ float cb1[NCODE * VEC];
  __shared__ float nrm0[NCODE];
  __shared__ float nrm1[NCODE];

  const int t = threadIdx.x;
  const int p = blockIdx.y;
  const int o = blockIdx.x * 256 + t;

  // Cooperative codebook staging: thread t loads code t of each level.
  {
    const float* c0 = codebooks + ((size_t)p * NCODE + t) * VEC;
    const float* c1 = codebooks + (((size_t)NPOS + p) * NCODE + t) * VEC;
    v4f a0 = *(const v4f*)(c0);
    v4f a1 = *(const v4f*)(c0 + 4);
    v4f b0 = *(const v4f*)(c1);
    v4f b1 = *(const v4f*)(c1 + 4);
    float n0 = 0.f, n1 = 0.f;
#pragma unroll
    for (int j = 0; j < 4; ++j) {
      cb0[t * VEC + j]     = a0[j];  n0 += a0[j] * a0[j];
      cb0[t * VEC + 4 + j] = a1[j];  n0 += a1[j] * a1[j];
      cb1[t * VEC + j]     = b0[j];  n1 += b0[j] * b0[j];
      cb1[t * VEC + 4 + j] = b1[j];  n1 += b1[j] * b1[j];
    }
    nrm0[t] = n0;
    nrm1[t] = n1;
  }
  __syncthreads();

  const int i0 = p * VEC;
  const float s = scales[(size_t)o * NGROUP + (i0 >> 7)];

  float w[VEC], r2[VEC], q[VEC];
#pragma unroll
  for (int j = 0; j < VEC; ++j) {
    w[j] = weight[(size_t)o * I_DIM + i0 + j] / s;
    r2[j] = -2.0f * w[j];        // residual pre-scaled for score FMAs
  }

  // --- level 0: argmin_k (||c||^2 - 2 r.c) -------------------------------
  int best = 0;
  float bestScore = 3.4e38f;
  for (int k = 0; k < NCODE; ++k) {
    float sc = nrm0[k];
#pragma unroll
    for (int j = 0; j < VEC; ++j) sc = fmaf(r2[j], cb0[k * VEC + j], sc);
    if (sc < bestScore) { bestScore = sc; best = k; }
  }
#pragma unroll
  for (int j = 0; j < VEC; ++j) {
    q[j] = cb0[best * VEC + j];
    r2[j] = -2.0f * (w[j] - q[j]);
  }

  // --- level 1 ------------------------------------------------------------
  best = 0;
  bestScore = 3.4e38f;
  for (int k = 0; k < NCODE; ++k) {
    float sc = nrm1[k];
#pragma unroll
    for (int j = 0; j < VEC; ++j) sc = fmaf(r2[j], cb1[k * VEC + j], sc);
    if (sc < bestScore) { bestScore = sc; best = k; }
  }
#pragma unroll
  for (int j = 0; j < VEC; ++j) q[j] += cb1[best * VEC + j];

  // Store transposed + rescaled; consecutive o across lanes -> coalesced.
#pragma unroll
  for (int j = 0; j < VEC; ++j)
    qwT[(size_t)(i0 + j) * O_DIM + o] = q[j] * s;
}

// ---------------------------------------------------------------------------
// Phase 2: Y[M,N] = X[M,K] * qwT[K,N] + bias[N] with fp32 WMMA 16x16x4.
// Block = 256 threads (8 wave32), 128x128 tile, waves 4(M) x 2(N),
// each wave: 2x4 accumulators (32x64).
//
// sX : [BM][BK+4]            -> A frag = contiguous K pair (ds_load_*_b64)
// sW2: [BK/2][2*BN+32]       -> K-pair interleaved: element (k,n) at
//                               [k/2][2n + (k&1)]; B frag = one b64 load,
//                               row stride 288 dwords => halves of the wave
//                               hit disjoint bank sets (0-31 vs 32-63).
// Register software pipeline: next tile's global loads issued before the
// 64-WMMA compute block; committed to LDS after the barrier.
// ---------------------------------------------------------------------------
#define BM 128
#define BN 128
#define BK 32
#define SX_STRIDE (BK + 4)         // 36
#define SW_STRIDE (2 * BN + 32)    // 288

__global__ __launch_bounds__(256) void gemm_wmma_f32_kernel(
    const float* __restrict__ X,     // [M, K]
    const float* __restrict__ Wt,    // [K, N]  (= qw^T)
    const float* __restrict__ bias,  // [N]
    float* __restrict__ Y)           // [M, N]
{
  __shared__ float sX[BM][SX_STRIDE];
  __shared__ float sW2[BK / 2][SW_STRIDE];

  const int tid  = threadIdx.x;
  const int wave = tid >> 5;
  const int lane = tid & 31;
  const int lm   = lane & 15;   // M (A) / N (B,D) within tile
  const int lh   = lane >> 4;   // half-wave selector

  const int m0 = blockIdx.y * BM;
  const int n0 = blockIdx.x * BN;
  const int wm = (wave & 3) << 5;   // 0,32,64,96
  const int wn = (wave >> 2) << 6;  // 0,64

  // Per-thread staging indices (fixed across K stages).
  const int xrow[4] = {tid >> 3, (256 + tid) >> 3, (512 + tid) >> 3, (768 + tid) >> 3};
  const int xc4     = (tid & 7) << 2;
  const int wrp[2]  = {tid >> 5, (256 + tid) >> 5};   // k-pair row 0..15
  const int wc4     = (tid & 31) << 2;

  v4f rx[4];          // X tile pipeline regs
  v4f rwa[2], rwb[2]; // W tile pipeline regs (even/odd K rows)

  auto load_tiles = [&](int kk) {
#pragma unroll
    for (int it = 0; it < 4; ++it)
      rx[it] = *(const v4f*)(X + (size_t)(m0 + xrow[it]) * I_DIM + kk + xc4);
#pragma unroll
    for (int it = 0; it < 2; ++it) {
      const float* base = Wt + (size_t)(kk + 2 * wrp[it]) * O_DIM + n0 + wc4;
      rwa[it] = *(const v4f*)(base);
      rwb[it] = *(const v4f*)(base + O_DIM);
    }
  };

  auto commit_tiles = [&]() {
#pragma unroll
    for (int it = 0; it < 4; ++it)
      *(v4f*)&sX[xrow[it]][xc4] = rx[it];
#pragma unroll
    for (int it = 0; it < 2; ++it) {
#pragma unroll
      for (int j = 0; j < 4; ++j) {
        v2f pr = {rwa[it][j], rwb[it][j]};
        *(v2f*)&sW2[wrp[it]][(wc4 + j) * 2] = pr;
      }
    }
  };

  v8f acc[2][4];
#pragma unroll
  for (int i = 0; i < 2; ++i)
#pragma unroll
    for (int j = 0; j < 4; ++j) acc[i][j] = v8f{};

  // Prologue: stage K-tile 0.
  load_tiles(0);
  commit_tiles();
  __syncthreads();

  for (int kk = 0; kk < I_DIM; kk += BK) {
    const bool more = (kk + BK) < I_DIM;
    if (more) load_tiles(kk + BK);   // in flight during compute

    // --- compute: 8 k-steps x 8 WMMAs ------------------------------------
#pragma unroll
    for (int ks = 0; ks < BK; ks += 4) {
      v2f a[2], b[4];
#pragma unroll
      for (int i = 0; i < 2; ++i) {
        // A 16x4 f32: lane -> M = lane%16, K = (lane/16)*2 + {0,1}
        const float* ap = &sX[wm + i * 16 + lm][ks + lh * 2];
        a[i].x = ap[0];
        a[i].y = ap[1];
      }
#pragma unroll
      for (int j = 0; j < 4; ++j) {
        // B 4x16 f32 (mirrored): single b64, K pair = ks/2 + lh
        b[j] = *(const v2f*)&sW2[(ks >> 1) + lh][(wn + j * 16 + lm) * 2];
      }
#pragma unroll
      for (int i = 0; i < 2; ++i)
#pragma unroll
        for (int j = 0; j < 4; ++j)
          acc[i][j] = __builtin_amdgcn_wmma_f32_16x16x4_f32(
              false, a[i], false, b[j], (short)0, acc[i][j], false, false);
    }

    if (more) {
      __syncthreads();      // compute done reading current tile
      commit_tiles();       // waits on loadcnt, writes next tile
      __syncthreads();      // tile visible to all waves
    }
  }

  // --- epilogue: D layout lane -> N = lane%16; VGPR v -> M = v + 8*(lane/16)
#pragma unroll
  for (int j = 0; j < 4; ++j) {
    const int n = n0 + wn + j * 16 + lm;
    const float bv = bias[n];
#pragma unroll
    for (int i = 0; i < 2; ++i) {
#pragma unroll
      for (int v = 0; v < 8; ++v) {
        const int m = m0 + wm + i * 16 + lh * 8 + v;
        Y[(size_t)m * O_DIM + n] = acc[i][j][v] + bv;
      }
    }
  }
}

// ---------------------------------------------------------------------------
extern "C" void kernel_launch(void* const* d_in, const int* in_sizes, int n_in,
                              void* d_out, int out_size, void* d_ws, size_t ws_size,
                              hipStream_t stream) {
  const float* x         = (const float*)d_in[0];  // [4,512,2048] -> [2048,2048]
  const float* weight    = (const float*)d_in[1];  // [2048,2048]
  const float* bias      = (const float*)d_in[2];  // [2048]
  const float* scales    = (const float*)d_in[3];  // [2048,16]
  const float* codebooks = (const float*)d_in[4];  // [2,256,256,8]
  float* y   = (float*)d_out;                      // [2048,2048]
  float* qwT = (float*)d_ws;                       // [2048,2048] scratch, 16 MB

  // Phase 1: quantize weights -> qwT (transposed, rescaled)
  rvq_quant_kernel<<<dim3(O_DIM / 256, NPOS), 256, 0, stream>>>(
      weight, scales, codebooks, qwT);

  // Phase 2: y = x @ qwT + bias via fp32 WMMA
  gemm_wmma_f32_kernel<<<dim3(O_DIM / BN, M_DIM / BM), 256, 0, stream>>>(
      x, qwT, bias, y);
}